// MNIST_RNN_15418932592980
// MI455X (gfx1250) — compile-verified
//
#include <hip/hip_runtime.h>

// ---------------------------------------------------------------------------
// Fused 2-layer LSTM (H=10, D=28, T=28) + linear head, fp32, WMMA 16x16x4.
// One wave = one 16-row batch tile. Gates padded to 4 N-tiles of 16 so the
// elementwise cell math is lane-local. Weights live in LDS as pre-permuted
// WMMA B-fragments (hoisted to VGPRs by the compiler); h is transposed
// D-frag -> A-frag through per-wave LDS. Branch-free elementwise via
// v_exp_f32 / v_rcp_f32; x loads are clamped instead of branch-guarded.
// ---------------------------------------------------------------------------

typedef float v2f __attribute__((ext_vector_type(2)));
typedef float v8f __attribute__((ext_vector_type(8)));

#define T_STEPS 28
#define D_IN    28
#define HID     10

// B-fragment table layout in LDS (fragment index -> [matrix][gate][k-chunk])
#define F_IH0 0            // w_ih0: 4 gates x 7 k-chunks (K=28)
#define F_HH0 28           // w_hh0: 4 gates x 3 k-chunks (K=10->12)
#define F_IH1 40           // w_ih1: 4 x 3
#define F_HH1 52           // w_hh1: 4 x 3
#define F_CLS 64           // w_cls: 1 tile x 3 k-chunks
#define NFRAG 67

// Branch-free activations: native v_exp_f32 + v_rcp_f32, saturate correctly.
__device__ __forceinline__ float sigf(float x) {
    return __builtin_amdgcn_rcpf(1.0f + __expf(-x));
}
__device__ __forceinline__ float tanhf_fast(float x) {
    // tanh(x) = 1 - 2/(exp(2x)+1);  x->+inf => 1, x->-inf => -1
    return 1.0f - 2.0f * __builtin_amdgcn_rcpf(1.0f + __expf(2.0f * x));
}

__device__ __forceinline__ v8f wmma4(v2f a, v2f b, v8f c) {
    // D = A(16x4 f32) * B(4x16 f32) + C(16x16 f32), exact fp32
    return __builtin_amdgcn_wmma_f32_16x16x4_f32(
        /*neg_a=*/false, a, /*neg_b=*/false, b,
        /*c_mod=*/(short)0, c, /*reuse_a=*/false, /*reuse_b=*/false);
}

__device__ __forceinline__ v2f ld_bfrag(const float2* wf, int lane) {
    float2 t = wf[lane];
    v2f b; b.x = t.x; b.y = t.y;
    return b;
}

__global__ __launch_bounds__(256) void lstm_fused_kernel(
    const float* __restrict__ x,
    const float* __restrict__ w_ih0, const float* __restrict__ w_hh0,
    const float* __restrict__ b_ih0, const float* __restrict__ b_hh0,
    const float* __restrict__ w_ih1, const float* __restrict__ w_hh1,
    const float* __restrict__ b_ih1, const float* __restrict__ b_hh1,
    const float* __restrict__ w_cls, const float* __restrict__ b_cls,
    float* __restrict__ out, int B)
{
    // Weight B-fragments: [frag][lane] = {k = 4c+2*(lane/16)+0, +1}, col n = lane%16
    __shared__ float2 wfrag[NFRAG][32];
    // Per-wave h transpose buffer: [wave][m][j], padded to 17 to dodge bank conflicts
    __shared__ float hbuf[8][16][17];

    const int tid = threadIdx.x;

    // ---- stage all weight fragments into LDS (done once, then loop-invariant) ----
    for (int idx = tid; idx < NFRAG * 64; idx += 256) {
        const int frag = idx >> 6;
        const int r    = idx & 63;
        const int ln   = r >> 1;
        const int v    = r & 1;
        const int n    = ln & 15;                 // N (column) of the tile
        const int kb   = ((ln >> 4) << 1) + v;    // 2*hi + v
        float val = 0.0f;
        if (frag < F_HH0) {                       // w_ih0 [40,28]
            const int g = frag / 7, c = frag % 7, k = 4 * c + kb;
            if (n < HID && k < D_IN) val = w_ih0[(g * HID + n) * D_IN + k];
        } else if (frag < F_IH1) {                // w_hh0 [40,10]
            const int f2 = frag - F_HH0, g = f2 / 3, c = f2 % 3, k = 4 * c + kb;
            if (n < HID && k < HID) val = w_hh0[(g * HID + n) * HID + k];
        } else if (frag < F_HH1) {                // w_ih1 [40,10]
            const int f2 = frag - F_IH1, g = f2 / 3, c = f2 % 3, k = 4 * c + kb;
            if (n < HID && k < HID) val = w_ih1[(g * HID + n) * HID + k];
        } else if (frag < F_CLS) {                // w_hh1 [40,10]
            const int f2 = frag - F_HH1, g = f2 / 3, c = f2 % 3, k = 4 * c + kb;
            if (n < HID && k < HID) val = w_hh1[(g * HID + n) * HID + k];
        } else {                                  // w_cls [10,10]
            const int c = frag - F_CLS, k = 4 * c + kb;
            if (n < HID && k < HID) val = w_cls[n * HID + k];
        }
        ((float*)&wfrag[frag][ln])[v] = val;
    }
    __syncthreads();

    const int wave = tid >> 5;
    const int lane = tid & 31;
    const int lo   = lane & 15;      // N column (accs) / M row (A-frags)
    const int hi   = lane >> 4;

    const int b0   = blockIdx.x * 128 + wave * 16;   // batch tile base
    const int brow = b0 + lo;                        // A-matrix row of this lane
    // Clamp instead of branch: rows past B re-read row B-1 (discarded at store).
    const int brow_c = (brow < B) ? brow : (B - 1);

    // Per-lane gate biases (broadcast over M; zero on padding columns)
    float bias0[4], bias1[4];
#pragma unroll
    for (int g = 0; g < 4; ++g) {
        bias0[g] = (lo < HID) ? (b_ih0[g * HID + lo] + b_hh0[g * HID + lo]) : 0.0f;
        bias1[g] = (lo < HID) ? (b_ih1[g * HID + lo] + b_hh1[g * HID + lo]) : 0.0f;
    }
    const float bcls = (lo < HID) ? b_cls[lo] : 0.0f;

    // Recurrent state
    v2f a_h0[3], a_h1[3];
#pragma unroll
    for (int c = 0; c < 3; ++c) {
        a_h0[c].x = 0.0f; a_h0[c].y = 0.0f;
        a_h1[c].x = 0.0f; a_h1[c].y = 0.0f;
    }
    float c0v[8], c1v[8];
#pragma unroll
    for (int v = 0; v < 8; ++v) { c0v[v] = 0.0f; c1v[v] = 0.0f; }

    const float* xrow = x + (size_t)brow_c * (T_STEPS * D_IN);

    for (int t = 0; t < T_STEPS; ++t) {
        // ---- x_t as A-fragments: lane holds row m=lo, k = 4c + 2*hi + {0,1} ----
        v2f xa[7];
#pragma unroll
        for (int c = 0; c < 7; ++c) {
            const int k = 4 * c + 2 * hi;
            const float2 p = *(const float2*)(xrow + t * D_IN + k);
            xa[c].x = p.x; xa[c].y = p.y;
        }
        if (t + 1 < T_STEPS)
            __builtin_prefetch(xrow + (t + 1) * D_IN, 0, 1);

        // ---- layer 0: gates = x_t @ W_ih0^T + h0 @ W_hh0^T + bias ----
        v8f acc0[4];
#pragma unroll
        for (int g = 0; g < 4; ++g) {
#pragma unroll
            for (int v = 0; v < 8; ++v) acc0[g][v] = bias0[g];
        }
#pragma unroll
        for (int c = 0; c < 7; ++c) {
#pragma unroll
            for (int g = 0; g < 4; ++g)
                acc0[g] = wmma4(xa[c], ld_bfrag(wfrag[F_IH0 + g * 7 + c], lane), acc0[g]);
        }
#pragma unroll
        for (int c = 0; c < 3; ++c) {
#pragma unroll
            for (int g = 0; g < 4; ++g)
                acc0[g] = wmma4(a_h0[c], ld_bfrag(wfrag[F_HH0 + g * 3 + c], lane), acc0[g]);
        }
        // LSTM cell, lane-local (padding columns stay exactly h=0)
        float h0v[8];
#pragma unroll
        for (int v = 0; v < 8; ++v) {
            const float ig = sigf(acc0[0][v]);
            const float fg = sigf(acc0[1][v]);
            const float gg = tanhf_fast(acc0[2][v]);
            const float og = sigf(acc0[3][v]);
            c0v[v] = fg * c0v[v] + ig * gg;
            h0v[v] = og * tanhf_fast(c0v[v]);
        }
        // D-frag -> A-frag transpose through per-wave LDS (same-wave DS is in-order)
#pragma unroll
        for (int v = 0; v < 8; ++v) hbuf[wave][v + 8 * hi][lo] = h0v[v];
#pragma unroll
        for (int c = 0; c < 3; ++c) {
            const int k = 4 * c + 2 * hi;     // k up to 11; hbuf[..][10..15] == 0
            a_h0[c].x = hbuf[wave][lo][k];
            a_h0[c].y = hbuf[wave][lo][k + 1];
        }

        // ---- layer 1: gates = h0_t @ W_ih1^T + h1 @ W_hh1^T + bias ----
        v8f acc1[4];
#pragma unroll
        for (int g = 0; g < 4; ++g) {
#pragma unroll
            for (int v = 0; v < 8; ++v) acc1[g][v] = bias1[g];
        }
#pragma unroll
        for (int c = 0; c < 3; ++c) {
#pragma unroll
            for (int g = 0; g < 4; ++g)
                acc1[g] = wmma4(a_h0[c], ld_bfrag(wfrag[F_IH1 + g * 3 + c], lane), acc1[g]);
        }
#pragma unroll
        for (int c = 0; c < 3; ++c) {
#pragma unroll
            for (int g = 0; g < 4; ++g)
                acc1[g] = wmma4(a_h1[c], ld_bfrag(wfrag[F_HH1 + g * 3 + c], lane), acc1[g]);
        }
        float h1v[8];
#pragma unroll
        for (int v = 0; v < 8; ++v) {
            const float ig = sigf(acc1[0][v]);
            const float fg = sigf(acc1[1][v]);
            const float gg = tanhf_fast(acc1[2][v]);
            const float og = sigf(acc1[3][v]);
            c1v[v] = fg * c1v[v] + ig * gg;
            h1v[v] = og * tanhf_fast(c1v[v]);
        }
#pragma unroll
        for (int v = 0; v < 8; ++v) hbuf[wave][v + 8 * hi][lo] = h1v[v];
#pragma unroll
        for (int c = 0; c < 3; ++c) {
            const int k = 4 * c + 2 * hi;
            a_h1[c].x = hbuf[wave][lo][k];
            a_h1[c].y = hbuf[wave][lo][k + 1];
        }
    }

    // ---- classifier: out = hT1 @ w_cls^T + b_cls ----
    v8f accc;
#pragma unroll
    for (int v = 0; v < 8; ++v) accc[v] = bcls;
#pragma unroll
    for (int c = 0; c < 3; ++c)
        accc = wmma4(a_h1[c], ld_bfrag(wfrag[F_CLS + c], lane), accc);

    if (lo < HID) {
#pragma unroll
        for (int v = 0; v < 8; ++v) {
            const int b = b0 + v + 8 * hi;
            if (b < B) out[b * 10 + lo] = accc[v];
        }
    }
}

extern "C" void kernel_launch(void* const* d_in, const int* in_sizes, int n_in,
                              void* d_out, int out_size, void* d_ws, size_t ws_size,
                              hipStream_t stream) {
    const float* x     = (const float*)d_in[0];
    const float* w_ih0 = (const float*)d_in[1];
    const float* w_hh0 = (const float*)d_in[2];
    const float* b_ih0 = (const float*)d_in[3];
    const float* b_hh0 = (const float*)d_in[4];
    const float* w_ih1 = (const float*)d_in[5];
    const float* w_hh1 = (const float*)d_in[6];
    const float* b_ih1 = (const float*)d_in[7];
    const float* b_hh1 = (const float*)d_in[8];
    const float* w_cls = (const float*)d_in[9];
    const float* b_cls = (const float*)d_in[10];

    const int B = in_sizes[0] / (T_STEPS * D_IN);   // 32768 for this problem
    const int blocks = (B + 127) / 128;             // 128 batch rows per block (8 waves x 16)

    lstm_fused_kernel<<<blocks, 256, 0, stream>>>(
        x, w_ih0, w_hh0, b_ih0, b_hh0,
        w_ih1, w_hh1, b_ih1, b_hh1,
        w_cls, b_cls, (float*)d_out, B);
}